// Sampling_22462678958130
// MI455X (gfx1250) — compile-verified
//
#include <hip/hip_runtime.h>
#include <hip/hip_bf16.h>

// CDNA5 / gfx1250, wave32.
typedef float v2f __attribute__((ext_vector_type(2)));
typedef float v4f __attribute__((ext_vector_type(4)));
typedef float v8f __attribute__((ext_vector_type(8)));

#define BN    512      // 8 * 16 * 4 (b,n) pairs, one workgroup each
#define HW    196      // rows per feature tile
#define CDIM  384      // channels
#define TK    98       // top-k = hw * 0.5
#define NWAVE 8        // 256 threads = 8 wave32

__global__ __launch_bounds__(256)
void Sampling_topk_attn_kernel(const float* __restrict__ token,
                               const float* __restrict__ feature,
                               float* __restrict__ out)
{
    const int bn   = blockIdx.x;          // 0..511
    const int tid  = threadIdx.x;         // 0..255
    const int lane = tid & 31;
    const int wave = tid >> 5;            // 0..7

    const float* feat = feature + (size_t)bn * (HW * CDIM);
    const float* tok  = token   + (size_t)bn * CDIM;
    float*       o    = out     + (size_t)bn * CDIM;

    __shared__ __align__(16) float s_tok[CDIM];
    __shared__ __align__(16) float s_w[HW];    // masked exp-weights
    __shared__ float s_logit[HW];

    // Hint the feature tile toward the caches (global_prefetch_b8).
    __builtin_prefetch(feat, 0, 1);

    // ---- Phase 0: token row -> LDS (384 = 3 * 128) ----
    if (tid < 128) {
        s_tok[tid]       = tok[tid];
        s_tok[tid + 128] = tok[tid + 128];
        s_tok[tid + 256] = tok[tid + 256];
    }
    __syncthreads();

    // ---- Phase 1: logits.  One row per wave iteration.  Each lane reads
    // 3 contiguous float4 chunks, so every load instruction is a 512 B/wave
    // global_load_b128 — this is the single HBM pass that sets the
    // 23.3 TB/s bandwidth floor (~6.6 us for the 154 MB feature tensor). ----
    const float scale = 0.05103103630798287f;   // 384^-0.5
    const v4f* tokv = (const v4f*)s_tok;
    for (int h = wave; h < HW; h += NWAVE) {
        const v4f* rowv = (const v4f*)(feat + (size_t)h * CDIM);
        float acc = 0.f;
        #pragma unroll
        for (int i = 0; i < CDIM / (4 * 32); ++i) {          // 3 iterations
            v4f f = rowv[lane + 32 * i];
            v4f t = tokv[lane + 32 * i];
            acc = fmaf(f.x, t.x, acc);
            acc = fmaf(f.y, t.y, acc);
            acc = fmaf(f.z, t.z, acc);
            acc = fmaf(f.w, t.w, acc);
        }
        #pragma unroll
        for (int off = 16; off > 0; off >>= 1)
            acc += __shfl_xor(acc, off, 32);
        if (lane == 0) s_logit[h] = acc * scale;
    }
    __syncthreads();

    // ---- Phase 1b: per-row rank + max (uniform LDS-broadcast reads), then
    // masked softmax numerator.  rank<98 == top-98 of the logits (softmax is
    // monotone); ties break toward lower index like lax.top_k.  The final
    // einsum is an order-independent sum over the selected set, so zeroing
    // non-selected weights replaces any gather/compaction. ----
    if (tid < HW) {
        const float x = s_logit[tid];
        float m = -3.4e38f;
        int cnt = 0;
        for (int j = 0; j < HW; ++j) {
            const float y = s_logit[j];
            m = fmaxf(m, y);
            cnt += ((y > x) || (y == x && j < tid)) ? 1 : 0;
        }
        s_w[tid] = (cnt < TK) ? __expf(x - m) : 0.f;
    }
    __syncthreads();

    // ---- Phase 1c: normalization constant (uniform broadcast loop). ----
    float snorm = 0.f;
    for (int j = 0; j < HW; ++j) snorm += s_w[j];
    const float inv = 1.f / snorm;

    // ---- Phase 2: out(1x384) = w(1x196) x F(196x384) via
    // V_WMMA_F32_16X16X4_F32.  ISA layout: A(16x4) lane m<16 holds row m
    // (K0,K1), lane m+16 holds row m (K2,K3); B(4x16) v0 = {lanes0-15:K0,
    // lanes16-31:K2}, v1 = {K1,K3}; D row M=0 = acc[0] on lanes 0-15.
    //
    // Key trick: D row m depends only on A row m, and we read only row 0 of
    // D — so ALL lanes can load the same weight pair unconditionally (a
    // conflict-free LDS broadcast, one aligned ds_load_b64).  Every D row
    // then just replicates row 0.  No EXEC masking, no cndmask in the loop.
    // Feature rows are re-read through L2 (154 MB tensor < 192 MB L2). ----
    const int half = lane >> 4;              // 0 | 1
    const int n    = lane & 15;              // column within c-tile

    for (int ct = wave; ct < CDIM / 16; ct += NWAVE) {   // 24 tiles, 3/wave
        const int c0 = ct * 16;
        v8f acc = {0.f, 0.f, 0.f, 0.f, 0.f, 0.f, 0.f, 0.f};
        for (int k4 = 0; k4 < HW / 4; ++k4) {            // 49 chunks * 4 = 196
            const int t0 = k4 * 4;
            // A: broadcast weight pair (K = t0+2*half, t0+2*half+1); 8B aligned.
            v2f A = *(const v2f*)&s_w[t0 + 2 * half];
            // B: two rows of the 4x16 feature tile for this lane's column.
            const float* r0 = feat + (size_t)(t0 + 2 * half) * CDIM + c0 + n;
            v2f B;
            B.x = r0[0];        // K = t0 + 2*half
            B.y = r0[CDIM];     // K = t0 + 2*half + 1
            acc = __builtin_amdgcn_wmma_f32_16x16x4_f32(
                /*neg_a=*/false, A, /*neg_b=*/false, B,
                /*c_mod=*/(short)0, acc, /*reuse_a=*/false, /*reuse_b=*/false);
        }
        if (half == 0) o[c0 + n] = acc[0] * inv;
    }
}

extern "C" void kernel_launch(void* const* d_in, const int* in_sizes, int n_in,
                              void* d_out, int out_size, void* d_ws, size_t ws_size,
                              hipStream_t stream) {
    const float* token   = (const float*)d_in[0];   // (8,16,4,384)
    const float* feature = (const float*)d_in[1];   // (8,16,4,196,384)
    float*       out     = (float*)d_out;           // (8,16,4,384)
    (void)in_sizes; (void)n_in; (void)out_size; (void)d_ws; (void)ws_size;

    Sampling_topk_attn_kernel<<<BN, 256, 0, stream>>>(token, feature, out);
}